// KANExpert_42683384988080
// MI455X (gfx1250) — compile-verified
//
#include <hip/hip_runtime.h>
#include <hip/hip_bf16.h>

typedef __attribute__((ext_vector_type(2))) float v2f;
typedef __attribute__((ext_vector_type(4))) float v4f;
typedef __attribute__((ext_vector_type(8))) float v8f;

#define BATCH   4096
#define IN_DIM  512
#define OUT_DIM 512
#define NB      8      // basis functions per input (GRID_SIZE + K)
#define BM      128    // block tile M
#define BN      64     // block tile N
#define BK      64     // K chunk = 8 inputs * 8 basis
#define IPC     8      // inputs per chunk
#define LDA     68     // LDS row stride (floats): 272B = 16B-aligned rows,
                       // conflict-free b64 fragment loads
#define LDB     68

// out[b,j] = sum_i sum_g basis(x[b,i])[g] * coeff[i,j,g] * scaling[i,j]
// fp32 WMMA GEMM: A[4096 x 4096] basis (computed on the fly from x),
// B[4096 x 512] = coeff*scaling fused on load. A tile row-major in LDS,
// B tile TRANSPOSED in LDS so each fragment is one ds_load_b64.
__global__ __launch_bounds__(256) void kan_wmma_f32(
    const float* __restrict__ x,        // [BATCH, IN_DIM]
    const float* __restrict__ coeff,    // [IN_DIM, OUT_DIM, NB]
    const float* __restrict__ scaling,  // [IN_DIM, OUT_DIM]
    float* __restrict__ out)            // [BATCH, OUT_DIM]
{
    __shared__ __align__(16) float As [BM][LDA];   // basis tile   [m][k]
    __shared__ __align__(16) float Bst[BN][LDB];   // weight tile  [n][k]  (transposed)

    const int tid  = threadIdx.x;
    const int wave = tid >> 5;
    const int lane = tid & 31;
    const int half = lane >> 4;   // 0: lanes 0-15, 1: lanes 16-31
    const int lm   = lane & 15;

    const int mBase = blockIdx.x * BM;
    const int jBase = blockIdx.y * BN;

    const int wm = (wave & 3) * 32;   // wave row offset (4 waves along M)
    const int wn = (wave >> 2) * 32;  // wave col offset (2 waves along N)

    v8f c00 = {}, c01 = {}, c10 = {}, c11 = {};

    for (int ii = 0; ii < IN_DIM; ii += IPC) {
        // ---- A tile: cubic B-spline basis, 128 rows x 8 inputs x 8 basis ----
        // Uniform grid h = 0.4 on [-1,1]; cell c in 0..4, u = local coord.
        // Nonzero basis indices c..c+3: (1-u)^3/6, (3u^3-6u^2+4)/6,
        // (-3u^3+3u^2+3u+1)/6, u^3/6.  Branchless -> EXEC stays all-ones.
        #pragma unroll
        for (int p = tid; p < BM * IPC; p += 256) {
            const int row = p >> 3;
            const int il  = p & 7;
            const float xv = x[(size_t)(mBase + row) * IN_DIM + (ii + il)];
            float t  = (xv + 1.0f) * 2.5f;
            float fc = floorf(t);
            fc = fminf(fmaxf(fc, 0.0f), 4.0f);
            const int   cell = (int)fc;
            const float u   = t - fc;
            const float u2  = u * u, u3 = u2 * u;
            const float omu = 1.0f - u;
            const float w0 = omu * omu * omu * (1.0f / 6.0f);
            const float w1 = (3.0f * u3 - 6.0f * u2 + 4.0f) * (1.0f / 6.0f);
            const float w2 = (-3.0f * u3 + 3.0f * u2 + 3.0f * u + 1.0f) * (1.0f / 6.0f);
            const float w3 = u3 * (1.0f / 6.0f);
            float v[NB];
            #pragma unroll
            for (int g = 0; g < NB; ++g) {          // cndmask selects, no branches
                float val = 0.0f;
                val = (g == cell    ) ? w0 : val;
                val = (g == cell + 1) ? w1 : val;
                val = (g == cell + 2) ? w2 : val;
                val = (g == cell + 3) ? w3 : val;
                v[g] = val;
            }
            v4f lo = { v[0], v[1], v[2], v[3] };
            v4f hi = { v[4], v[5], v[6], v[7] };
            *(v4f*)&As[row][il * NB]     = lo;       // two ds_store_b128
            *(v4f*)&As[row][il * NB + 4] = hi;
        }
        // ---- B tile (transposed): Bst[n][k] = coeff[i,j,g]*scaling[i,j] ----
        // 4 contiguous g share one (i,j): 16B global load + 16B LDS store.
        #pragma unroll
        for (int p = tid; p < (BK * BN) / 4; p += 256) {
            const int n  = p >> 4;                  // 0..63
            const int kq = (p & 15) * 4;            // 0,4,...,60
            const int i  = ii + (kq >> 3);
            const int g0 = kq & 7;                  // 0 or 4
            const int j  = jBase + n;
            const v4f cv = *(const v4f*)&coeff[((size_t)i * OUT_DIM + j) * NB + g0];
            const float s = scaling[(size_t)i * OUT_DIM + j];
            v4f w = { cv.x * s, cv.y * s, cv.z * s, cv.w * s };
            *(v4f*)&Bst[n][kq] = w;
        }
        __syncthreads();

        // ---- 16 K-steps; per step: 4x ds_load_b64 -> 4x WMMA (4 chains) ----
        #pragma unroll
        for (int ks = 0; ks < BK; ks += 4) {
            const int k0 = ks + 2 * half;           // lanes 0-15: K={ks,ks+1}; 16-31: +2
            const v2f a0 = *(const v2f*)&As [wm + lm     ][k0];
            const v2f a1 = *(const v2f*)&As [wm + 16 + lm][k0];
            const v2f b0 = *(const v2f*)&Bst[wn + lm     ][k0];
            const v2f b1 = *(const v2f*)&Bst[wn + 16 + lm][k0];
            c00 = __builtin_amdgcn_wmma_f32_16x16x4_f32(false, a0, false, b0, (short)0, c00, false, false);
            c01 = __builtin_amdgcn_wmma_f32_16x16x4_f32(false, a0, false, b1, (short)0, c01, false, false);
            c10 = __builtin_amdgcn_wmma_f32_16x16x4_f32(false, a1, false, b0, (short)0, c10, false, false);
            c11 = __builtin_amdgcn_wmma_f32_16x16x4_f32(false, a1, false, b1, (short)0, c11, false, false);
        }
        __syncthreads();
    }

    // ---- epilogue: C/D layout — VGPR v holds M = v + 8*half, N = lm ----
    #pragma unroll
    for (int v = 0; v < 8; ++v) {
        const int m0 = mBase + wm + v + 8 * half;
        const int m1 = m0 + 16;
        out[(size_t)m0 * OUT_DIM + jBase + wn + lm]      = c00[v];
        out[(size_t)m0 * OUT_DIM + jBase + wn + 16 + lm] = c01[v];
        out[(size_t)m1 * OUT_DIM + jBase + wn + lm]      = c10[v];
        out[(size_t)m1 * OUT_DIM + jBase + wn + 16 + lm] = c11[v];
    }
}

extern "C" void kernel_launch(void* const* d_in, const int* in_sizes, int n_in,
                              void* d_out, int out_size, void* d_ws, size_t ws_size,
                              hipStream_t stream) {
    const float* x       = (const float*)d_in[0];   // [4096, 512]
    const float* coeff   = (const float*)d_in[1];   // [512, 512, 8]
    const float* scaling = (const float*)d_in[2];   // [512, 512]
    float* out = (float*)d_out;                     // [4096, 512]

    dim3 grid(BATCH / BM, OUT_DIM / BN);            // 32 x 8
    dim3 block(256);
    kan_wmma_f32<<<grid, block, 0, stream>>>(x, coeff, scaling, out);
}